// STA_36867999268899
// MI455X (gfx1250) — compile-verified
//
#include <hip/hip_runtime.h>

typedef __attribute__((ext_vector_type(2))) float v2f;
typedef __attribute__((ext_vector_type(8))) float v8f;
typedef unsigned int uv4 __attribute__((ext_vector_type(4)));
typedef int          iv4 __attribute__((ext_vector_type(4)));
typedef int          iv8 __attribute__((ext_vector_type(8)));

#define B_  8
#define T_  12
#define N_  250
#define NP  256
#define D_  64
#define H_  8
#define HD  8
#define W_  3
#define JW  (W_ * T_)   // 36
#define EPS_ 1e-5f

// LDS layout for attn_kernel (dynamic shared, byte offsets are descriptor constants)
#define LDS_K_OFF   0u                 // ks[256][8]  : 8192 B
#define LDS_V_OFF   8192u              // vs[256][8]  : 8192 B
#define LDS_Q_OFF   16384u             // qs[16][8]   : 512 B
#define LDS_S_OFF   16896u             // S[16][256]  : 16384 B
#define LDS_BYTES   33280u

#if __has_builtin(__builtin_amdgcn_tensor_load_to_lds) && __has_builtin(__builtin_amdgcn_s_wait_tensorcnt)
#define HAVE_TDM 1
#else
#define HAVE_TDM 0
#endif

#if HAVE_TDM
// Issue one TDM descriptor: contiguous 1-D tile of `nelem` f32 from global -> LDS.
// D# per CDNA5 ISA ch.8: group0 = {count=1 | lds_addr | global_addr(57b) | type=2},
// group1 = {data_size=4B, tensor_dim0=nelem, tensor_dim1=1, tile_dim0=nelem (1-D)}.
__device__ __forceinline__ void tdm_load_1d(unsigned lds_off, const float* gptr,
                                            unsigned nelem) {
  unsigned long long ga = (unsigned long long)(uintptr_t)gptr;
  uv4 g0;
  g0.x = 0x1u;                                            // count=1 (valid, user mode)
  g0.y = lds_off;                                         // lds_addr (bytes)
  g0.z = (unsigned)ga;                                    // global_addr[31:0]
  g0.w = ((unsigned)(ga >> 32) & 0x01ffffffu) | 0x80000000u; // addr[56:32] | type=2
  iv8 g1;
  g1.s0 = (int)(0x2u << 16);                              // data_size = 4 bytes
  g1.s1 = (int)((nelem & 0xffffu) << 16);                 // tensor_dim0[15:0]
  g1.s2 = (int)(((nelem >> 16) & 0xffffu) | (1u << 16));  // tensor_dim0[31:16], tensor_dim1=1
  g1.s3 = (int)((nelem & 0xffffu) << 16);                 // tile_dim0 = nelem (<= 65535)
  g1.s4 = 0;                                              // tile_dim1=0, tile_dim2=0 (1-D)
  g1.s5 = (int)nelem;                                     // tensor_dim0_stride[31:0]
  g1.s6 = 0;
  g1.s7 = 0;
  iv4 z4 = {0, 0, 0, 0};
#if defined(__clang_major__) && (__clang_major__ >= 23)
  iv8 z8 = {0, 0, 0, 0, 0, 0, 0, 0};
  __builtin_amdgcn_tensor_load_to_lds(g0, g1, z4, z4, z8, 0);
#else
  __builtin_amdgcn_tensor_load_to_lds(g0, g1, z4, z4, 0);
#endif
}
#endif

// ---------------- Stage 1: QKV projection ----------------
// grid: B*T*NP blocks, 64 threads. Pads node dim to 256 with zeros.
__global__ __launch_bounds__(64) void qkv_kernel(
    const float* __restrict__ x,
    const float* __restrict__ Wq, const float* __restrict__ bq,
    const float* __restrict__ Wk, const float* __restrict__ bk,
    const float* __restrict__ Wv, const float* __restrict__ bv,
    float* __restrict__ q_ws, float* __restrict__ k_ws, float* __restrict__ v_ws) {
  int bid = blockIdx.x;
  int n = bid % NP;
  int t = (bid / NP) % T_;
  int b = bid / (NP * T_);
  int d = threadIdx.x;          // 0..63
  int h = d / HD, hi = d % HD;
  size_t o = (((size_t)(b * T_ + t) * H_ + h) * NP + n) * HD + hi;
  if (n >= N_) {                // zero padding rows (whole block uniform)
    q_ws[o] = 0.f; k_ws[o] = 0.f; v_ws[o] = 0.f;
    return;
  }
  __shared__ float xr[D_];
  xr[d] = x[((size_t)(b * T_ + t) * N_ + n) * D_ + d];
  __syncthreads();
  float aq = bq[d], ak = bk[d], av = bv[d];
  for (int i = 0; i < D_; ++i) {
    float xv = xr[i];
    aq = fmaf(xv, Wq[d * D_ + i], aq);
    ak = fmaf(xv, Wk[d * D_ + i], ak);
    av = fmaf(xv, Wv[d * D_ + i], av);
  }
  q_ws[o] = aq; k_ws[o] = ak; v_ws[o] = av;
}

// ---------------- Stage 2: windowed attention, one wave32 per 16-row tile ----------------
// grid: B*T*H*W_*16 blocks of 32 threads, LDS_BYTES dynamic LDS.
__global__ __launch_bounds__(32) void attn_kernel(
    const float* __restrict__ q_ws, const float* __restrict__ k_ws,
    const float* __restrict__ v_ws, const float* __restrict__ adj,
    float* __restrict__ hs_ws) {
  extern __shared__ float smem[];
  float (*ks)[HD] = (float (*)[HD])(smem + LDS_K_OFF / 4);
  float (*vs)[HD] = (float (*)[HD])(smem + LDS_V_OFF / 4);
  float (*qs)[HD] = (float (*)[HD])(smem + LDS_Q_OFF / 4);
  float (*S)[NP]  = (float (*)[NP])(smem + LDS_S_OFF / 4);

  int bid  = blockIdx.x;
  int tile = bid & 15;
  int ti   = (bid >> 4) % W_;
  int h    = (bid / 48) % H_;
  int t    = (bid / 384) % T_;
  int b    = bid / 4608;
  int lane = threadIdx.x;

  // cumulative shifts/scales from the reference's in-loop reassignment
  int   tkv    = (ti == 2) ? t : (t + T_ - 1) % T_;
  int   tadj   = (ti == 0) ? (t + T_ - 1) % T_ : ((ti == 1) ? t : (t + 1) % T_);
  float qscale = (ti == 0) ? 0.35355339059f : ((ti == 1) ? 0.125f : 0.04419417382f);

  size_t qbase = ((size_t)(b * T_ + t)   * H_ + h) * NP * HD;
  size_t kbase = ((size_t)(b * T_ + tkv) * H_ + h) * NP * HD;
  int m0 = tile * 16;

#if HAVE_TDM
  // Tensor Data Mover: one descriptor per tile, wave-level issue, TENSORcnt tracked.
  tdm_load_1d(LDS_K_OFF, k_ws + kbase, NP * HD);
  tdm_load_1d(LDS_V_OFF, v_ws + kbase, NP * HD);
  tdm_load_1d(LDS_Q_OFF, q_ws + qbase + (size_t)m0 * HD, 16 * HD);
  __builtin_amdgcn_s_wait_tensorcnt(0);
#else
  for (int idx = lane; idx < NP * HD; idx += 32) {
    ks[idx / HD][idx % HD] = k_ws[kbase + idx];
    vs[idx / HD][idx % HD] = v_ws[kbase + idx];
  }
  for (int idx = lane; idx < 16 * HD; idx += 32)
    qs[idx / HD][idx % HD] = q_ws[qbase + (size_t)m0 * HD + idx];
#endif
  __syncthreads();

  int half = lane >> 4;   // 0: K pair {0,1}; 1: K pair {2,3} (ISA f32 A/B layout)
  int lq   = lane & 15;

  // ---- scores: S[16][256] = (qscale*Q)(16x8) * K^T(8x256), 2 WMMA per 16-col tile ----
  for (int jt = 0; jt < 16; ++jt) {
    int n0 = jt * 16;
    v8f c = {};
    for (int kb = 0; kb < HD; kb += 4) {
      int k0 = kb + half * 2;
      v2f a, bf;
      a.x  = qs[lq][k0] * qscale;  a.y  = qs[lq][k0 + 1] * qscale;
      bf.x = ks[n0 + lq][k0];      bf.y = ks[n0 + lq][k0 + 1];
      c = __builtin_amdgcn_wmma_f32_16x16x4_f32(false, a, false, bf,
                                                (short)0, c, false, false);
    }
    for (int r = 0; r < 8; ++r)          // C layout: VGPR r -> M=r (+8 upper half)
      S[r + half * 8][n0 + lq] = c[r];
  }
  __syncthreads();

  // ---- softmax over the real 250 columns, then * adj; zero padding ----
  if (lane < 16) {
    int m = m0 + lane;
    if (m < N_) {
      const float* arow = adj + ((size_t)(b * T_ + tadj) * N_ + m) * N_;
      __builtin_prefetch(arow, 0, 1);            // global_prefetch_b8 into L2/L0
      float mx = -3.4e38f;
      for (int n = 0; n < N_; ++n) mx = fmaxf(mx, S[lane][n]);
      float sum = 0.f;
      for (int n = 0; n < N_; ++n) { float e = __expf(S[lane][n] - mx); S[lane][n] = e; sum += e; }
      float inv = 1.f / sum;
      for (int n = 0; n < N_; ++n) S[lane][n] *= inv * arow[n];
      for (int n = N_; n < NP; ++n) S[lane][n] = 0.f;
    } else {
      for (int n = 0; n < NP; ++n) S[lane][n] = 0.f;
    }
  }
  __syncthreads();

  // ---- R = A(16x256) * V(256x8, padded to 16 cols): 64 WMMAs over K ----
  v8f c = {};
  for (int kb = 0; kb < NP; kb += 4) {
    int k0 = kb + half * 2;
    v2f a, bf;
    a.x  = S[lq][k0];  a.y = S[lq][k0 + 1];
    bf.x = (lq < HD) ? vs[k0][lq]     : 0.f;
    bf.y = (lq < HD) ? vs[k0 + 1][lq] : 0.f;
    c = __builtin_amdgcn_wmma_f32_16x16x4_f32(false, a, false, bf,
                                              (short)0, c, false, false);
  }

  // scatter into hs_ws[b][node][d][j], j = ti*T + t
  if (lq < HD) {
    int j = ti * T_ + t;
    int d = h * HD + lq;
    for (int r = 0; r < 8; ++r) {
      int m = m0 + r + half * 8;
      if (m < N_)
        hs_ws[((size_t)(b * N_ + m) * D_ + d) * JW + j] = c[r];
    }
  }
}

// ---------------- Stage 3: temporal mix + residual + LayerNorm ----------------
// grid: B*T*N blocks, 64 threads (one per channel d).
__global__ __launch_bounds__(64) void mix_ln_kernel(
    const float* __restrict__ hs_ws, const float* __restrict__ x,
    const float* __restrict__ Wd, const float* __restrict__ bd,
    const float* __restrict__ gamma, const float* __restrict__ beta,
    float* __restrict__ out) {
  int bid = blockIdx.x;
  int n = bid % N_;
  int t = (bid / N_) % T_;
  int b = bid / (N_ * T_);
  int d = threadIdx.x;

  const float* hrow = hs_ws + ((size_t)(b * N_ + n) * D_ + d) * JW;
  const float* wrow = Wd + t * JW;
  float acc = bd[t];
  for (int j = 0; j < JW; ++j) acc = fmaf(hrow[j], wrow[j], acc);
  float y = acc + x[((size_t)(b * T_ + t) * N_ + n) * D_ + d];

  __shared__ float red[D_];
  red[d] = y;
  __syncthreads();
  for (int s = 32; s > 0; s >>= 1) { if (d < s) red[d] += red[d + s]; __syncthreads(); }
  float mu = red[0] * (1.f / D_);
  __syncthreads();
  float dv = y - mu;
  red[d] = dv * dv;
  __syncthreads();
  for (int s = 32; s > 0; s >>= 1) { if (d < s) red[d] += red[d + s]; __syncthreads(); }
  float var = red[0] * (1.f / D_);
  float rstd = rsqrtf(var + EPS_);
  out[((size_t)(b * T_ + t) * N_ + n) * D_ + d] = gamma[d] * dv * rstd + beta[d];
}

extern "C" void kernel_launch(void* const* d_in, const int* in_sizes, int n_in,
                              void* d_out, int out_size, void* d_ws, size_t ws_size,
                              hipStream_t stream) {
  const float* x     = (const float*)d_in[0];
  // d_in[1] = ste (unused by the reference)
  const float* adj   = (const float*)d_in[2];
  const float* Wq    = (const float*)d_in[3];
  const float* bq    = (const float*)d_in[4];
  const float* Wk    = (const float*)d_in[5];
  const float* bk    = (const float*)d_in[6];
  const float* Wv    = (const float*)d_in[7];
  const float* bv    = (const float*)d_in[8];
  const float* Wd    = (const float*)d_in[9];
  const float* bd    = (const float*)d_in[10];
  const float* gamma = (const float*)d_in[11];
  const float* beta  = (const float*)d_in[12];
  float* out = (float*)d_out;

  float* ws = (float*)d_ws;
  const size_t qkv_elems = (size_t)B_ * T_ * H_ * NP * HD;   // 1,572,864 each
  float* q_ws  = ws;
  float* k_ws  = q_ws + qkv_elems;
  float* v_ws  = k_ws + qkv_elems;
  float* hs_ws = v_ws + qkv_elems;                           // B*N*D*36 = 4,608,000

  qkv_kernel<<<B_ * T_ * NP, 64, 0, stream>>>(x, Wq, bq, Wk, bk, Wv, bv,
                                              q_ws, k_ws, v_ws);
  attn_kernel<<<B_ * T_ * H_ * W_ * 16, 32, LDS_BYTES, stream>>>(q_ws, k_ws, v_ws,
                                                                 adj, hs_ws);
  mix_ln_kernel<<<B_ * T_ * N_, 64, 0, stream>>>(hs_ws, x, Wd, bd, gamma, beta, out);
}